// BGNN_50680614092915
// MI455X (gfx1250) — compile-verified
//
#include <hip/hip_runtime.h>
#include <math.h>

// fp32 WMMA fragment types for V_WMMA_F32_16X16X4_F32
typedef float v2f __attribute__((ext_vector_type(2)));
typedef float v8f __attribute__((ext_vector_type(8)));

#define KC     32   // K chunk staged in LDS per iteration
#define BM     128  // block tile M
#define BN     64   // block tile N
#define LDA_S  36   // padded LDS row stride (floats): 36r mod 64 distinct over 16 rows; 144B keeps align
#define LDB2_S 160  // k-pair interleaved B tile: row stride (floats); 160 mod 64 == 32 -> halves disjoint

// ---------------------------------------------------------------------------
// d[i] = rsqrt(max(sum_j adj[i,j], 1))   (PyG DenseGCNConv normalization)
// ---------------------------------------------------------------------------
__global__ __launch_bounds__(256) void deg_rsqrt_kernel(const float* __restrict__ adj,
                                                        float* __restrict__ d, int n) {
    __shared__ float red[256];
    const int row = blockIdx.x;
    const float* arow = adj + (size_t)row * n;
    float s = 0.f;
    for (int j = threadIdx.x; j < n; j += 256) s += arow[j];
    red[threadIdx.x] = s;
    __syncthreads();
    for (int off = 128; off > 0; off >>= 1) {
        if (threadIdx.x < off) red[threadIdx.x] += red[threadIdx.x + off];
        __syncthreads();
    }
    if (threadIdx.x == 0) {
        float deg = red[0] < 1.0f ? 1.0f : red[0];
        d[row] = rsqrtf(deg);
    }
}

// ---------------------------------------------------------------------------
// C[M,N] = epi( A[M,K] @ B[K,N] ), epi: v = rowScale[i]*acc + bias[j]; relu opt
// Row-major. N must be a multiple of BN (512/256 are). 256 threads = 8 wave32;
// wave (wm,wn) in 4x2 grid owns a 32x32 sub-tile (2x2 WMMA 16x16 tiles).
// Double-buffered LDS pipeline, one barrier per chunk, next-chunk globals
// staged in registers during the 32 WMMAs. B tile stored k-pair interleaved:
// element (k,col) at (k>>1)*LDB2_S + 2*col + (k&1)  => each B fragment is one
// aligned ds_load_b64 (no repack movs), conflict-free across lane halves.
// ---------------------------------------------------------------------------
__global__ __launch_bounds__(256) void wmma_gemm_f32(
    const float* __restrict__ A, const float* __restrict__ B,
    float* __restrict__ C, int M, int N, int K,
    const float* __restrict__ rowScale,   // length M, or nullptr
    const float* __restrict__ bias,       // length N, or nullptr
    int relu)
{
    __shared__ float lA[2][BM * LDA_S];        // 2 x 18432 B
    __shared__ float lB[2][(KC / 2) * LDB2_S]; // 2 x 10240 B

    const int tid   = threadIdx.x;
    const int lane  = tid & 31;
    const int wave  = tid >> 5;
    const int wm    = wave >> 1;          // 0..3 -> M offset 32*wm
    const int wn    = wave & 1;           // 0..1 -> N offset 32*wn
    const int m0    = blockIdx.y * BM;
    const int n0    = blockIdx.x * BN;
    const int lhalf = lane >> 4;          // 0/1
    const int lidx  = lane & 15;
    const bool fullM = (m0 + BM <= M);

    // fill-role indices (fixed per thread)
    const int ar_f  = tid >> 1;           // A: row 0..127
    const int acb_f = (tid & 1) * 16;     // A: col base 0 or 16
    const int kp_f  = tid >> 4;           // B: k-pair 0..15  (k rows 2kp, 2kp+1)
    const int bcb_f = (tid & 15) * 4;     // B: col base 0..60

    float aReg[16];
    float bReg[8];   // [0..3] = row 2kp cols c..c+3, [4..7] = row 2kp+1

    auto loadA = [&](int k0) {
        const int gr = m0 + ar_f;
        if (fullM && (k0 + KC <= K)) {
            const float* src = A + (size_t)gr * K + k0 + acb_f;
            #pragma unroll
            for (int c = 0; c < 16; ++c) aReg[c] = src[c];
        } else {
            const int grc = (gr < M) ? gr : 0;
            const float* src = A + (size_t)grc * K;
            #pragma unroll
            for (int c = 0; c < 16; ++c) {
                const int gk = k0 + acb_f + c;
                const bool ok = (gr < M) && (gk < K);
                const float v = src[ok ? gk : 0];
                aReg[c] = ok ? v : 0.f;
            }
        }
    };
    auto loadB = [&](int k0) {
        const int gk0 = k0 + 2 * kp_f;
        const int gk1 = gk0 + 1;
        if (k0 + KC <= K) {
            const float* s0 = B + (size_t)gk0 * N + n0 + bcb_f;
            const float* s1 = s0 + N;
            #pragma unroll
            for (int c = 0; c < 4; ++c) { bReg[c] = s0[c]; bReg[4 + c] = s1[c]; }
        } else {
            const bool ok0 = (gk0 < K);
            const bool ok1 = (gk1 < K);
            const float* s0 = B + (size_t)(ok0 ? gk0 : 0) * N + n0 + bcb_f;
            const float* s1 = B + (size_t)(ok1 ? gk1 : 0) * N + n0 + bcb_f;
            #pragma unroll
            for (int c = 0; c < 4; ++c) {
                const float v0 = s0[c];
                const float v1 = s1[c];
                bReg[c]     = ok0 ? v0 : 0.f;
                bReg[4 + c] = ok1 ? v1 : 0.f;
            }
        }
    };
    auto storeTiles = [&](int buf) {
        float* dstA = lA[buf] + ar_f * LDA_S + acb_f;
        #pragma unroll
        for (int c = 0; c < 16; ++c) dstA[c] = aReg[c];
        float* dstB = lB[buf] + kp_f * LDB2_S + 2 * bcb_f;
        #pragma unroll
        for (int c = 0; c < 4; ++c) {       // adjacent pair -> merged ds_store_b64
            dstB[2 * c]     = bReg[c];
            dstB[2 * c + 1] = bReg[4 + c];
        }
    };

    v8f acc[2][2] = {};

    const int nChunks = (K + KC - 1) / KC;

    // prologue: stage chunk 0
    loadA(0);
    loadB(0);
    storeTiles(0);
    __syncthreads();

    int cur = 0;
    for (int ci = 0; ci < nChunks; ++ci) {
        const int k0n = (ci + 1) * KC;
        const bool hasNext = (k0n < K);

        // issue next chunk's global loads early (latency hidden behind WMMAs)
        if (hasNext) { loadA(k0n); loadB(k0n); }

        const float* __restrict__ cA = lA[cur];
        const float* __restrict__ cB = lB[cur];

        #pragma unroll
        for (int kk = 0; kk < KC; kk += 4) {
            v2f afrag[2], bfrag[2];
            #pragma unroll
            for (int tm = 0; tm < 2; ++tm) {
                const int ar = 32 * wm + 16 * tm + lidx;
                const int ak = kk + 2 * lhalf;
                afrag[tm].x = cA[ar * LDA_S + ak];
                afrag[tm].y = cA[ar * LDA_S + ak + 1];
            }
            #pragma unroll
            for (int tn = 0; tn < 2; ++tn) {
                const int bc = 32 * wn + 16 * tn + lidx;
                const int bk = kk + 2 * lhalf;              // even
                const float* p = cB + (bk >> 1) * LDB2_S + 2 * bc;
                bfrag[tn].x = p[0];                         // one ds_load_b64
                bfrag[tn].y = p[1];
            }
            #pragma unroll
            for (int tm = 0; tm < 2; ++tm)
                #pragma unroll
                for (int tn = 0; tn < 2; ++tn)
                    acc[tm][tn] = __builtin_amdgcn_wmma_f32_16x16x4_f32(
                        /*neg_a=*/false, afrag[tm], /*neg_b=*/false, bfrag[tn],
                        /*c_mod=*/(short)0, acc[tm][tn],
                        /*reuse_a=*/false, /*reuse_b=*/false);
        }

        // park next chunk into the alternate buffer after current reads
        if (hasNext) storeTiles(cur ^ 1);
        __syncthreads();
        cur ^= 1;
    }

    // ---- epilogue: C = [relu](rowScale*acc + bias); D layout M = r + 8*lhalf ----
    #pragma unroll
    for (int tm = 0; tm < 2; ++tm) {
        #pragma unroll
        for (int tn = 0; tn < 2; ++tn) {
            const int gcol = n0 + 32 * wn + 16 * tn + lidx;
            const float bv = bias ? bias[gcol] : 0.f;
            if (fullM) {
                #pragma unroll
                for (int r = 0; r < 8; ++r) {
                    const int grow = m0 + 32 * wm + 16 * tm + r + 8 * lhalf;
                    const float s = rowScale ? rowScale[grow] : 1.0f;
                    float v = s * acc[tm][tn][r] + bv;
                    if (relu) v = v > 0.f ? v : 0.f;
                    C[(size_t)grow * N + gcol] = v;
                }
            } else {
                #pragma unroll
                for (int r = 0; r < 8; ++r) {
                    const int grow = m0 + 32 * wm + 16 * tm + r + 8 * lhalf;
                    if (grow < M) {
                        const float s = rowScale ? rowScale[grow] : 1.0f;
                        float v = s * acc[tm][tn][r] + bv;
                        if (relu) v = v > 0.f ? v : 0.f;
                        C[(size_t)grow * N + gcol] = v;
                    }
                }
            }
        }
    }
}

// ---------------------------------------------------------------------------
// Launch: deg -> T1 = d.*(x@W1) -> H1 = relu(d.*(adj@T1)+b1)
//             -> T2 = d.*(H1@W2) -> out = relu(d.*(adj@T2)+b2)
// ---------------------------------------------------------------------------
extern "C" void kernel_launch(void* const* d_in, const int* in_sizes, int n_in,
                              void* d_out, int out_size, void* d_ws, size_t ws_size,
                              hipStream_t stream) {
    const float* x   = (const float*)d_in[0];
    const float* adj = (const float*)d_in[1];
    const float* W1  = (const float*)d_in[2];
    const float* b1  = (const float*)d_in[3];
    const float* W2  = (const float*)d_in[4];
    const float* b2  = (const float*)d_in[5];

    const int H1DIM = in_sizes[3];                 // 512
    const int H2DIM = in_sizes[5];                 // 256
    const int N     = in_sizes[2] / H1DIM;         // 2339

    char* ws = (char*)d_ws;
    float* dvec = (float*)ws;  ws += ((size_t)N * sizeof(float) + 255) & ~(size_t)255;
    float* T1   = (float*)ws;  ws += (size_t)N * H1DIM * sizeof(float);
    float* H1   = (float*)ws;  ws += (size_t)N * H1DIM * sizeof(float);
    float* T2   = (float*)ws;

    deg_rsqrt_kernel<<<N, 256, 0, stream>>>(adj, dvec, N);

    const dim3 blk(256);
    const dim3 g1(H1DIM / BN, (N + BM - 1) / BM);
    const dim3 g2(H2DIM / BN, (N + BM - 1) / BM);

    // T1 = d[:,None] * (x @ W1)
    wmma_gemm_f32<<<g1, blk, 0, stream>>>(x,   W1, T1, N, H1DIM, N,     dvec, nullptr, 0);
    // H1 = relu(d[:,None] * (adj @ T1) + b1)
    wmma_gemm_f32<<<g1, blk, 0, stream>>>(adj, T1, H1, N, H1DIM, N,     dvec, b1,      1);
    // T2 = d[:,None] * (H1 @ W2)
    wmma_gemm_f32<<<g2, blk, 0, stream>>>(H1,  W2, T2, N, H2DIM, H1DIM, dvec, nullptr, 0);
    // out = relu(d[:,None] * (adj @ T2) + b2)
    wmma_gemm_f32<<<g2, blk, 0, stream>>>(adj, T2, (float*)d_out, N, H2DIM, N, dvec, b2, 1);
}